// Router_85298050499170
// MI455X (gfx1250) — compile-verified
//
#include <hip/hip_runtime.h>
#include <hip/hip_bf16.h>
#include <math.h>

typedef __attribute__((ext_vector_type(2))) float v2f;
typedef __attribute__((ext_vector_type(4))) float v4f;
typedef __attribute__((ext_vector_type(8))) float v8f;

#define N_EXP 8

// ---------------------------------------------------------------------------
// Kernel 1: streaming zero-fill of the dense output (the bandwidth hog:
// ~335 MB @ 23.3 TB/s ~= 14 us). Non-temporal 128-bit stores.
// ---------------------------------------------------------------------------
__global__ void router_zero_kernel(float* __restrict__ out, long long n) {
    long long n4 = n >> 2;
    v4f z = {0.f, 0.f, 0.f, 0.f};
    long long stride = (long long)gridDim.x * blockDim.x;
    for (long long i = (long long)blockIdx.x * blockDim.x + threadIdx.x;
         i < n4; i += stride) {
        __builtin_nontemporal_store(z, ((v4f*)out) + i);
    }
    if (blockIdx.x == 0 && threadIdx.x < (n & 3))
        out[(n4 << 2) + threadIdx.x] = 0.f;
}

// ---------------------------------------------------------------------------
// Kernel 2: logits = x @ w_g^T via V_WMMA_F32_16X16X4_F32 (exact fp32).
// One wave per 16-token tile; N dimension holds the 8 experts (cols 8..15
// zero-padded with a lane select, so no divergence around the WMMA and
// EXEC stays all-ones as required).
//
// A layout (32-bit A 16x4): lanes 0-15 -> {K=0 (v0), K=1 (v1)},
//                           lanes 16-31 -> {K=2 (v0), K=3 (v1)}, M = lane&15.
// B layout (mirrors C row striping): v0 -> {K=0 | K=2}, v1 -> {K=1 | K=3},
//                           N = lane&15.
// D layout (32-bit C/D 16x16): VGPR v: lanes 0-15 M=v, lanes 16-31 M=v+8,
//                              N = lane&15.
// ---------------------------------------------------------------------------
__global__ void __launch_bounds__(256)
router_logits_wmma_kernel(const float* __restrict__ x,
                          const float* __restrict__ wg,
                          float* __restrict__ logits,
                          int D, int ntiles) {
    int tile = blockIdx.x * (blockDim.x >> 5) + (threadIdx.x >> 5);
    if (tile >= ntiles) return;                 // wave-uniform exit
    int lane  = threadIdx.x & 31;
    int half  = lane >> 4;                      // 0: K=0,1   1: K=2,3
    int col   = lane & 15;                      // A: row M / B: col N
    bool evalid = (col < N_EXP);
    int ecl   = col & (N_EXP - 1);

    const float* xrow = x  + ((long long)tile * 16 + col) * D + half * 2;
    const float* wrow = wg + (long long)ecl * D + half * 2;

    v8f acc = {};
    #pragma unroll 4
    for (int k = 0; k < D; k += 4) {
        v2f a, b;
        a.x = xrow[k];
        a.y = xrow[k + 1];
        float bx = wrow[k];
        float by = wrow[k + 1];
        b.x = evalid ? bx : 0.f;                // lane select, no branch
        b.y = evalid ? by : 0.f;
        acc = __builtin_amdgcn_wmma_f32_16x16x4_f32(
            /*neg_a=*/false, a, /*neg_b=*/false, b,
            /*c_mod=*/(short)0, acc, /*reuse_a=*/false, /*reuse_b=*/false);
    }

    if (evalid) {
        int rowbase = tile * 16 + half * 8;
        #pragma unroll
        for (int v = 0; v < 8; v++)
            logits[(long long)(rowbase + v) * N_EXP + col] = acc[v];
    }
}

// ---------------------------------------------------------------------------
// Kernel 3: per-token softmax over 8 logits + top-2 (stable tie-break: lower
// index wins, matching jax.lax.top_k). One thread per token.
// Layout in ws: idx[i] / p[i] indexed by flattened entry i = k*N + n.
// ---------------------------------------------------------------------------
__global__ void router_topk_softmax_kernel(const float* __restrict__ logits,
                                           int* __restrict__ idx_out,
                                           float* __restrict__ p_out, int N) {
    int n = blockIdx.x * blockDim.x + threadIdx.x;
    if (n >= N) return;
    float l[N_EXP];
    float mx = -INFINITY;
    #pragma unroll
    for (int e = 0; e < N_EXP; e++) {
        l[e] = logits[(long long)n * N_EXP + e];
        mx = fmaxf(mx, l[e]);
    }
    float p[N_EXP];
    float sum = 0.f;
    #pragma unroll
    for (int e = 0; e < N_EXP; e++) { p[e] = expf(l[e] - mx); sum += p[e]; }
    float inv = 1.f / sum;

    int i0 = 0; float b0 = l[0];
    #pragma unroll
    for (int e = 1; e < N_EXP; e++)
        if (l[e] > b0) { b0 = l[e]; i0 = e; }
    int i1 = -1; float b1 = -INFINITY;
    #pragma unroll
    for (int e = 0; e < N_EXP; e++)
        if (e != i0 && l[e] > b1) { b1 = l[e]; i1 = e; }

    idx_out[n]     = i0;
    idx_out[N + n] = i1;
    p_out[n]       = p[i0] * inv;
    p_out[N + n]   = p[i1] * inv;
}

// ---------------------------------------------------------------------------
// Kernel 4: deterministic ordered rank scan over the 2N flattened entries
// (exactly the reference cumsum order), capacity drop, scatter into cb_weight
// and sec_mask, and used_capacity. Single workgroup (8 wave32 waves); per-
// chunk per-expert prefix counts via __ballot + popcount, cross-wave via LDS.
// Only 8192 entries -> microseconds.
// ---------------------------------------------------------------------------
__global__ void __launch_bounds__(256)
router_scan_scatter_kernel(const int* __restrict__ idx,
                           const float* __restrict__ p,
                           float* __restrict__ out, int N, int C) {
    const int total = 2 * N;
    const long long NEC = (long long)N * N_EXP * C;
    float* cb  = out + N_EXP;
    float* sec = out + N_EXP + NEC;

    __shared__ int wave_tot[8][N_EXP];

    int tid  = threadIdx.x;
    int lane = tid & 31;
    int wv   = tid >> 5;
    unsigned lanemask_lt = (1u << lane) - 1u;

    int base[N_EXP];
    #pragma unroll
    for (int e = 0; e < N_EXP; e++) base[e] = 0;

    for (int chunk = 0; chunk < total; chunk += 256) {
        int i = chunk + tid;
        int e = (i < total) ? idx[i] : -1;

        unsigned bal[N_EXP];
        #pragma unroll
        for (int ee = 0; ee < N_EXP; ee++)
            bal[ee] = (unsigned)__ballot(e == ee);

        if (lane < N_EXP)
            wave_tot[wv][lane] = __popc(bal[lane]);
        __syncthreads();

        if (e >= 0) {
            int pre = __popc(bal[e] & lanemask_lt);
            int cross = 0;
            for (int w = 0; w < wv; w++) cross += wave_tot[w][e];
            int rank = base[e] + cross + pre;
            if (rank < C) {
                float wval = p[i];
                int n = (i < N) ? i : (i - N);
                long long off = (long long)n * (N_EXP * C)
                              + (long long)e * C + rank;
                cb[off]  = wval;
                sec[off] = (wval != 0.f) ? 1.f : 0.f;
            }
        }

        #pragma unroll
        for (int ee = 0; ee < N_EXP; ee++) {
            int s = 0;
            #pragma unroll
            for (int w = 0; w < 8; w++) s += wave_tot[w][ee];
            base[ee] += s;
        }
        __syncthreads();
    }

    if (tid < N_EXP)
        out[tid] = (float)((base[tid] < C) ? base[tid] : C);
}

// ---------------------------------------------------------------------------
extern "C" void kernel_launch(void* const* d_in, const int* in_sizes, int n_in,
                              void* d_out, int out_size, void* d_ws, size_t ws_size,
                              hipStream_t stream) {
    const float* x  = (const float*)d_in[0];   // [B,T,D] fp32
    const float* wg = (const float*)d_in[1];   // [E,D]   fp32
    float* out = (float*)d_out;

    int D = (n_in >= 2) ? (in_sizes[1] / N_EXP) : 1024;
    int N = in_sizes[0] / D;                   // B*T tokens

    // capacity (matches _capacity in the reference)
    int C = (int)floor(2.0 * 1.25 * (double)N / (double)N_EXP);
    C += (C & 1);
    if (C < 4) C = 4;

    // workspace layout
    float* ws_logits = (float*)d_ws;                                   // N*8 f32
    int*   ws_idx    = (int*)  ((char*)d_ws + (size_t)N * N_EXP * 4);  // 2N  i32
    float* ws_p      = (float*)((char*)d_ws + (size_t)N * N_EXP * 4
                                            + (size_t)2 * N * 4);      // 2N  f32

    // 1) zero the dense output (bandwidth-dominant step)
    router_zero_kernel<<<2048, 256, 0, stream>>>(out, (long long)out_size);

    // 2) logits GEMM via fp32 WMMA (8 tiles of 16 tokens per 256-thread block)
    int ntiles = (N + 15) / 16;
    int blocks = (ntiles + 7) / 8;
    router_logits_wmma_kernel<<<blocks, 256, 0, stream>>>(x, wg, ws_logits, D, ntiles);

    // 3) per-token softmax + top-2
    router_topk_softmax_kernel<<<(N + 255) / 256, 256, 0, stream>>>(
        ws_logits, ws_idx, ws_p, N);

    // 4) ordered rank scan + capacity drop + scatter (single workgroup)
    router_scan_scatter_kernel<<<1, 256, 0, stream>>>(ws_idx, ws_p, out, N, C);
}